// PixelUnShuffle_13666585936262
// MI455X (gfx1250) — compile-verified
//
#include <hip/hip_runtime.h>

// PixelUnshuffle / depth-to-space, s=2, in [16,128,128,256] f32 -> out [16,256,256,64].
// Pure data movement: 512 MiB total traffic, roofline ~23us @ 23.3 TB/s.
// Strategy: async global->LDS staging (ASYNCcnt path), padded-LDS transpose,
// non-temporal b128 stores. No WMMA: 0 FLOPs, matrix ops would only add traffic.

typedef float v4f __attribute__((ext_vector_type(4)));
typedef int   v4i __attribute__((ext_vector_type(4)));
typedef __attribute__((address_space(3))) v4i lds_v4i;

// Tile = 16 input pixels = 16*256 floats = 16KB. LDS padded +16B per 64B -> 20KB.
// Chunk (16B) index mapping: ci -> ci + (ci>>2).

__global__ __launch_bounds__(256) void pixel_unshuffle_kernel(
    const float* __restrict__ in, float* __restrict__ out) {
  __shared__ __align__(16) float smem[5120];  // 20KB padded tile

  const int t    = threadIdx.x;       // 0..255
  const int tile = blockIdx.x;        // 16 pixels per tile
  const int tileFloatBase = tile * 4096;

  // ---------------- stage 16KB input tile into LDS ----------------
#if __has_builtin(__builtin_amdgcn_global_load_async_to_lds_b128)
  {
#pragma unroll
    for (int q = 0; q < 4; ++q) {
      const int ci  = t + 256 * q;        // 16B chunk index in tile
      const int cip = ci + (ci >> 2);     // padded chunk index
      v4i*     g = (v4i*)(in + tileFloatBase + ci * 4);
      lds_v4i* l = (lds_v4i*)(smem + cip * 4);
      // lanes consecutive in t -> 512B contiguous per wave per issue
      __builtin_amdgcn_global_load_async_to_lds_b128(g, l, 0, 0);
    }
  }
#if __has_builtin(__builtin_amdgcn_s_wait_asynccnt)
  __builtin_amdgcn_s_wait_asynccnt(0);
#else
  asm volatile("s_wait_asynccnt 0" ::: "memory");
#endif
#else
  // Fallback: regular coalesced b128 loads, ds_store_b128 into LDS
#pragma unroll
  for (int q = 0; q < 4; ++q) {
    const int ci  = t + 256 * q;
    const int cip = ci + (ci >> 2);
    v4f v = __builtin_nontemporal_load((const v4f*)(in + tileFloatBase + ci * 4));
    *(v4f*)(smem + cip * 4) = v;
  }
#endif
  __syncthreads();

  // ---------------- transpose + store ----------------
  // thread t: pl = pixel-in-tile (0..15), cc = channel-group (0..15)
  const int pl = t >> 4;
  const int cc = t & 15;
  const int p  = tile * 16 + pl;      // global pixel index, < 16*128*128
  const int b  = p >> 14;             // / (128*128)
  const int h  = (p >> 7) & 127;
  const int w  = p & 127;

  // read 4 float4s from this pixel at 64B stride (padded -> conflict-free)
  // Li[m] = in[b,h,w, 16cc + 4i + m],   m = k*2 + j,   c = 4cc + i
  v4f L[4];
#pragma unroll
  for (int i = 0; i < 4; ++i) {
    const int ci  = pl * 64 + cc * 4 + i;
    const int cip = ci + (ci >> 2);     // = pl*80 + cc*5 + i
    L[i] = *(const v4f*)(smem + cip * 4);
  }

  // out[b, 2h+j, 2w+k, 4cc + r] = L[r][k*2+j]
  const int ohBase = (b << 8) + 2 * h;  // b*256 + 2h
#pragma unroll
  for (int m = 0; m < 4; ++m) {
    const int j = m & 1;
    const int k = m >> 1;
    const int oidx = (((ohBase + j) << 8) + (2 * w + k)) * 64 + (cc << 2);
    v4f o = { L[0][m], L[1][m], L[2][m], L[3][m] };
    // 16 lanes (cc=0..15) -> one contiguous 256B channel block per position
    __builtin_nontemporal_store(o, (v4f*)(out + oidx));
  }
}

extern "C" void kernel_launch(void* const* d_in, const int* in_sizes, int n_in,
                              void* d_out, int out_size, void* d_ws, size_t ws_size,
                              hipStream_t stream) {
  const float* in = (const float*)d_in[0];
  float* out      = (float*)d_out;

  const int n_floats = in_sizes[0];        // 16*128*128*256 = 67108864
  const int blocks   = n_floats / 4096;    // 16 pixels (4096 floats) per block
  pixel_unshuffle_kernel<<<dim3(blocks), dim3(256), 0, stream>>>(in, out);
}